// StealNMSLoss_678604833526
// MI455X (gfx1250) — compile-verified
//
#include <hip/hip_runtime.h>
#include <stdint.h>
#include <math.h>

// ---------------- problem constants ----------------
#define IMG_H 1024
#define IMG_W 1024
#define NBATCH 8
#define TILE_W 64
#define TILE_H 32
#define NTHREADS 256

// extended LDS tile sizes
#define TR_H (TILE_H + 4)   // 36   true tile rows (+2 halo each side)
#define TR_WP 72            // physical row stride; logical col c -> phys c+2
                            // => interior (logical col 2) at byte offset row*288+16 (16B aligned)
#define EP_H (TILE_H + 4)   // 36   exp(pred) rows (vertical 5-tap)
#define EP_W (TILE_W)       // 64
#define DX_H (TILE_H + 2)   // 34   first-derivative tiles
#define DX_W (TILE_W + 2)   // 66

#define NBLK_X (IMG_W / TILE_W)   // 16
#define NBLK_Y (IMG_H / TILE_H)   // 32
#define NPARTIALS (NBLK_X * NBLK_Y * NBATCH)  // 4096

// ---------------- CDNA5 async global->LDS path ----------------
#if defined(__has_builtin)
#define KHAS(x) __has_builtin(x)
#else
#define KHAS(x) 0
#endif

#if KHAS(__builtin_amdgcn_global_load_async_to_lds_b32)
#define USE_ASYNC_LDS 1
#else
#define USE_ASYNC_LDS 0
#endif

#if USE_ASYNC_LDS && KHAS(__builtin_amdgcn_global_load_async_to_lds_b128)
#define USE_ASYNC_B128 1
#else
#define USE_ASYNC_B128 0
#endif

#define AS1 __attribute__((address_space(1)))
#define AS3 __attribute__((address_space(3)))

typedef int v4i_gcc __attribute__((vector_size(4 * sizeof(int))));

__device__ __forceinline__ void async_copy_f32(const float* g, float* l) {
#if USE_ASYNC_LDS
  __builtin_amdgcn_global_load_async_to_lds_b32(
      (AS1 int*)(uintptr_t)g, (AS3 int*)(uint32_t)(uintptr_t)l,
      /*offset=*/0, /*cpol=*/0);
#else
  *l = *g;
#endif
}

__device__ __forceinline__ void async_copy_f32x4(const float* g, float* l) {
#if USE_ASYNC_B128
  __builtin_amdgcn_global_load_async_to_lds_b128(
      (AS1 v4i_gcc*)(uintptr_t)g, (AS3 v4i_gcc*)(uint32_t)(uintptr_t)l,
      /*offset=*/0, /*cpol=*/0);
#else
  async_copy_f32(g + 0, l + 0);
  async_copy_f32(g + 1, l + 1);
  async_copy_f32(g + 2, l + 2);
  async_copy_f32(g + 3, l + 3);
#endif
}

__device__ __forceinline__ void wait_async_all() {
#if USE_ASYNC_LDS
#if KHAS(__builtin_amdgcn_s_wait_asynccnt)
  __builtin_amdgcn_s_wait_asynccnt(0);
#else
  asm volatile("s_wait_asynccnt 0x0" ::: "memory");
#endif
#endif
}

__device__ __forceinline__ int clampi(int v, int lo, int hi) {
  return v < lo ? lo : (v > hi ? hi : v);
}

__device__ __forceinline__ int reflect_row(int gy) {
  gy = (gy < 0) ? -gy : gy;
  return (gy > IMG_H - 1) ? (2 * (IMG_H - 1) - gy) : gy;
}

// ---------------- main fused kernel ----------------
__global__ __launch_bounds__(NTHREADS) void steal_nms_main(
    const float* __restrict__ pred, const float* __restrict__ truev,
    float* __restrict__ partials) {
  __shared__ __attribute__((aligned(16))) float s_true[TR_H * TR_WP];
  __shared__ __attribute__((aligned(16))) float s_ep[EP_H * EP_W];
  __shared__ float s_dx[DX_H * DX_W];
  __shared__ float s_dy[DX_H * DX_W];
  __shared__ float s_red[NTHREADS];

  const int tid = threadIdx.x;
  const int x0 = blockIdx.x * TILE_W;
  const int y0 = blockIdx.y * TILE_H;
  const float* tb = truev + (size_t)blockIdx.z * IMG_H * IMG_W;
  const float* pb = pred + (size_t)blockIdx.z * IMG_H * IMG_W;

  // logical col c of the true tile lives at s_true[row*TR_WP + c + 2]
#define ST(r, c) s_true[(r) * TR_WP + (c) + 2]

  // ---- phase 1: async tile loads into LDS ----
  // (a) true interior: logical cols 2..65 == global cols x0..x0+63, always
  // in-bounds horizontally; rows clamped (stage-1 'edge' pad). 16B aligned
  // on both sides -> B128 per lane.
  for (int e4 = tid; e4 < TR_H * (TILE_W / 4); e4 += NTHREADS) {
    const int r = e4 >> 4, lx4 = e4 & 15;            // row 0..35, group 0..15
    const int gy = clampi(y0 - 2 + r, 0, IMG_H - 1);
    async_copy_f32x4(tb + (size_t)gy * IMG_W + x0 + lx4 * 4, &ST(r, 2 + lx4 * 4));
  }
  // (b) true halo columns: logical cols {0,1,66,67} with horizontal clamp.
  {
    const int e = tid;  // 144 elements < NTHREADS: single masked pass
    if (e < TR_H * 4) {
      const int r = e >> 2, k = e & 3;
      const int c = (k < 2) ? k : (k + 64);          // 0,1,66,67
      const int gy = clampi(y0 - 2 + r, 0, IMG_H - 1);
      const int gx = clampi(x0 - 2 + c, 0, IMG_W - 1);
      async_copy_f32(tb + (size_t)gy * IMG_W + gx, &ST(r, c));
    }
  }
  // (c) pred rows with vertical reflect indices (kornia filter2D 'reflect');
  // rows are 256B aligned both sides -> B128 per lane.
  for (int e4 = tid; e4 < EP_H * (EP_W / 4); e4 += NTHREADS) {
    const int r = e4 >> 4, lx4 = e4 & 15;
    const int gy = reflect_row(y0 - 2 + r);
    async_copy_f32x4(pb + (size_t)gy * IMG_W + x0 + lx4 * 4, &s_ep[r * EP_W + lx4 * 4]);
  }
  wait_async_all();
  __syncthreads();

  // ---- phase 2: exp(pred / 0.1) in place ----
  for (int e = tid; e < EP_H * EP_W; e += NTHREADS)
    s_ep[e] = expf(10.0f * s_ep[e]);
  __syncthreads();

  // ---- phase 3: first sobel derivatives dx,dy on extended region ----
  // s_dx/s_dy[l] holds d at global row clamp(y0-1+l), col clamp(x0-1+c).
  for (int e = tid; e < DX_H * DX_W; e += NTHREADS) {
    const int ey = e / DX_W, ex = e - ey * DX_W;
    const int a = clampi(y0 - 1 + ey, 0, IMG_H - 1);
    const int b = clampi(x0 - 1 + ex, 0, IMG_W - 1);
    // second 'edge' clamp, then map to logical s_true cols (base x0-2)
    const int rm = clampi(a - 1, 0, IMG_H - 1) - (y0 - 2);
    const int r0 = a - (y0 - 2);
    const int rp = clampi(a + 1, 0, IMG_H - 1) - (y0 - 2);
    const int cm = clampi(b - 1, 0, IMG_W - 1) - (x0 - 2);
    const int c0 = b - (x0 - 2);
    const int cp = clampi(b + 1, 0, IMG_W - 1) - (x0 - 2);
    const float t_mm = ST(rm, cm);
    const float t_m0 = ST(rm, c0);
    const float t_mp = ST(rm, cp);
    const float t_0m = ST(r0, cm);
    const float t_0p = ST(r0, cp);
    const float t_pm = ST(rp, cm);
    const float t_p0 = ST(rp, c0);
    const float t_pp = ST(rp, cp);
    s_dx[e] = ((t_mp - t_mm) + 2.0f * (t_0p - t_0m) + (t_pp - t_pm)) * 0.125f;
    s_dy[e] = ((t_pm - t_mm) + 2.0f * (t_p0 - t_m0) + (t_pp - t_mp)) * 0.125f;
  }
  __syncthreads();

  // ---- phase 4: per-pixel loss ----
  float acc = 0.0f;
  const int lx = tid & (TILE_W - 1);
  const int lyb = tid >> 6;  // 0..3
  const int gx = x0 + lx;
  const int cm = clampi(gx - 1, 0, IMG_W - 1) - (x0 - 1);
  const int c0 = lx + 1;
  const int cp = clampi(gx + 1, 0, IMG_W - 1) - (x0 - 1);
#pragma unroll
  for (int r = 0; r < TILE_H / 4; ++r) {
    const int ly = lyb + 4 * r;
    const int gy = y0 + ly;
    // vertical 5-tap normalization (reflect handled at load)
    const float e0 = s_ep[(ly + 0) * EP_W + lx];
    const float e1 = s_ep[(ly + 1) * EP_W + lx];
    const float e2 = s_ep[(ly + 2) * EP_W + lx];
    const float e3 = s_ep[(ly + 3) * EP_W + lx];
    const float e4 = s_ep[(ly + 4) * EP_W + lx];
    const float resp = (e0 + e1 + e2 + e3 + e4) + 1e-6f;
    float ratio = e2 / resp;
    ratio = fminf(fmaxf(ratio, 1e-6f), 1.0f);
    const float log_norm = logf(ratio);

    // second-stage sobel indices (stage-2 'edge' clamp)
    const int rm = clampi(gy - 1, 0, IMG_H - 1) - (y0 - 1);
    const int r0 = ly + 1;
    const int rp = clampi(gy + 1, 0, IMG_H - 1) - (y0 - 1);

    const float xx = ((s_dx[rm * DX_W + cp] - s_dx[rm * DX_W + cm]) +
                      2.0f * (s_dx[r0 * DX_W + cp] - s_dx[r0 * DX_W + cm]) +
                      (s_dx[rp * DX_W + cp] - s_dx[rp * DX_W + cm])) * 0.125f;
    const float xy = ((s_dy[rm * DX_W + cp] - s_dy[rm * DX_W + cm]) +
                      2.0f * (s_dy[r0 * DX_W + cp] - s_dy[r0 * DX_W + cm]) +
                      (s_dy[rp * DX_W + cp] - s_dy[rp * DX_W + cm])) * 0.125f;
    const float yy = ((s_dy[rp * DX_W + cm] - s_dy[rm * DX_W + cm]) +
                      2.0f * (s_dy[rp * DX_W + c0] - s_dy[rm * DX_W + c0]) +
                      (s_dy[rp * DX_W + cp] - s_dy[rm * DX_W + cp])) * 0.125f;

    const float sarg = -xy + 1e-7f;
    const float sgn = (sarg > 0.0f) ? 1.0f : ((sarg < 0.0f) ? -1.0f : 0.0f);
    const float theta = atanf(yy * sgn / (xx + 1e-7f));
    float ang = theta * 57.29577951308232f;  // rad -> deg
    ang = (ang < 0.0f) ? (ang + 180.0f) : ang;

    float w = 0.0f;  // the 4 disjoint section masks (NaN angle -> 0)
    w += ((ang >= 0.0f && ang < 22.5f) || (ang >= 157.5f && ang < 180.0f)) ? 1.0f : 0.0f;
    w += (ang >= 22.5f && ang < 67.5f) ? 1.0f : 0.0f;
    w += (ang >= 112.5f && ang < 157.5f) ? 1.0f : 0.0f;
    w += (ang >= 67.5f && ang < 112.5f) ? 1.0f : 0.0f;

    acc += log_norm * w;
  }

  // ---- deterministic block reduction ----
  s_red[tid] = acc;
  __syncthreads();
#pragma unroll
  for (int s = NTHREADS / 2; s > 0; s >>= 1) {
    if (tid < s) s_red[tid] += s_red[tid + s];
    __syncthreads();
  }
  if (tid == 0) {
    const int pidx = ((int)blockIdx.z * NBLK_Y + (int)blockIdx.y) * NBLK_X + (int)blockIdx.x;
    partials[pidx] = s_red[0];
  }
#undef ST
}

// ---------------- final deterministic reduction ----------------
__global__ __launch_bounds__(NTHREADS) void steal_nms_reduce(
    const float* __restrict__ partials, float* __restrict__ out) {
  __shared__ double red[NTHREADS];
  double s = 0.0;
  for (int i = threadIdx.x; i < NPARTIALS; i += NTHREADS) s += (double)partials[i];
  red[threadIdx.x] = s;
  __syncthreads();
#pragma unroll
  for (int k = NTHREADS / 2; k > 0; k >>= 1) {
    if (threadIdx.x < k) red[threadIdx.x] += red[threadIdx.x + k];
    __syncthreads();
  }
  if (threadIdx.x == 0) {
    const double denom = 4.0 * (double)NBATCH * (double)IMG_H * (double)IMG_W;
    out[0] = (float)(-red[0] / denom);
  }
}

// ---------------- entry point ----------------
extern "C" void kernel_launch(void* const* d_in, const int* in_sizes, int n_in,
                              void* d_out, int out_size, void* d_ws, size_t ws_size,
                              hipStream_t stream) {
  (void)in_sizes; (void)n_in; (void)out_size; (void)ws_size;
  const float* pred = (const float*)d_in[0];   // pred_labels [8,1,1024,1024] f32
  const float* truev = (const float*)d_in[1];  // true_labels [8,1,1024,1024] f32
  float* out = (float*)d_out;                  // scalar f32
  float* partials = (float*)d_ws;              // NPARTIALS floats of scratch

  dim3 grid(NBLK_X, NBLK_Y, NBATCH);
  steal_nms_main<<<grid, NTHREADS, 0, stream>>>(pred, truev, partials);
  steal_nms_reduce<<<1, NTHREADS, 0, stream>>>(partials, out);
}